// BaseMoeModule_1065151889873
// MI455X (gfx1250) — compile-verified
//
#include <hip/hip_runtime.h>
#include <hip/hip_bf16.h>
#include <stdint.h>

#define N_TOK   8192
#define HID     2048
#define N_EXP   64
#define TOPK    8
#define N_ITEM  (N_TOK * TOPK)    // 65536 replicated items
#define HB      256               // items per histogram block
#define NHB     (N_ITEM / HB)     // 256 histogram blocks
#define NPB     (N_TOK / 8)       // 1024 routing blocks (8 tokens each)

typedef float v4f __attribute__((ext_vector_type(4)));
typedef int   v4i __attribute__((ext_vector_type(4)));

__device__ __forceinline__ void amax2(float& v, int& i, float ov, int oi) {
    // max with tie -> lower index (matches jax.lax.top_k ordering)
    if (ov > v || (ov == v && oi < i)) { v = ov; i = oi; }
}

// ---------------------------------------------------------------------------
// Kernel 1: softmax over 64 experts, top-8 (stable ties), weight renorm.
// One wave32 per token (8 tokens / 256-thread block). Deterministic per-block
// reduction of router probabilities for the LBL "p" term (no float atomics).
// ---------------------------------------------------------------------------
__global__ void k_route(const float* __restrict__ logits,
                        float* __restrict__ sel_w,
                        int*   __restrict__ sel_idx,
                        float* __restrict__ pblk) {
    __shared__ float pw[8][N_EXP];
    const int tid  = threadIdx.x;
    const int lane = tid & 31;
    const int wv   = tid >> 5;
    const int token = blockIdx.x * 8 + wv;

    const float* lr = logits + (size_t)token * N_EXP;
    float l0 = lr[lane];
    float l1 = lr[lane + 32];

    // wave-wide softmax over 64 entries (2 per lane)
    float m = fmaxf(l0, l1);
    for (int off = 16; off; off >>= 1) m = fmaxf(m, __shfl_xor(m, off, 32));
    float e0 = expf(l0 - m), e1 = expf(l1 - m);
    float s = e0 + e1;
    for (int off = 16; off; off >>= 1) s += __shfl_xor(s, off, 32);
    float p0 = e0 / s, p1 = e1 / s;

    pw[wv][lane]      = p0;
    pw[wv][lane + 32] = p1;

    // top-8 by repeated wave argmax (tie -> lower expert index)
    float q0 = p0, q1 = p1;
    float selv = 0.f, ssum = 0.f;
    int   seli = 0;
    for (int k = 0; k < TOPK; ++k) {
        float v = q0; int i = lane;
        amax2(v, i, q1, lane + 32);
        for (int off = 16; off; off >>= 1) {
            float ov = __shfl_xor(v, off, 32);
            int   oi = __shfl_xor(i, off, 32);
            amax2(v, i, ov, oi);
        }
        ssum += v;
        if (lane == k) { selv = v; seli = i; }
        if (i == lane)       q0 = -1.f;   // probs are > 0, -1 = removed
        if (i == lane + 32)  q1 = -1.f;
    }
    if (lane < TOPK) {
        sel_w[token * TOPK + lane]   = selv / ssum;
        sel_idx[token * TOPK + lane] = seli;
    }

    // deterministic per-block reduction of probs (fixed order over 8 waves)
    __syncthreads();
    if (tid < N_EXP) {
        float acc = 0.f;
        #pragma unroll
        for (int w = 0; w < 8; ++w) acc += pw[w][tid];
        pblk[(size_t)blockIdx.x * N_EXP + tid] = acc;
    }
}

// ---------------------------------------------------------------------------
// Kernel 2: per-block expert histograms + stable local ranks.
// ---------------------------------------------------------------------------
__global__ void k_hist(const int* __restrict__ sel_idx,
                       int* __restrict__ block_hist,
                       int* __restrict__ lrank) {
    __shared__ int hist[N_EXP];
    __shared__ int ex[HB];
    const int t = threadIdx.x, b = blockIdx.x;
    const int i = b * HB + t;
    const int e = sel_idx[i];
    ex[t] = e;
    if (t < N_EXP) hist[t] = 0;
    __syncthreads();
    atomicAdd(&hist[e], 1);            // integer count: order-independent
    int r = 0;
    for (int j = 0; j < t; ++j) r += (ex[j] == e);   // stable local rank
    lrank[i] = r;
    __syncthreads();
    if (t < N_EXP) block_hist[b * N_EXP + t] = hist[t];
}

// ---------------------------------------------------------------------------
// Kernel 3: exclusive scans -> stable sort bases, group_sizes, LBL loss.
// Single block, 64 threads (one per expert).
// ---------------------------------------------------------------------------
__global__ void k_scan(int* __restrict__ block_hist,
                       const float* __restrict__ pblk,
                       int* __restrict__ eoff,
                       int* __restrict__ gsz,
                       float* __restrict__ out_gs,
                       float* __restrict__ out_lb) {
    __shared__ float pt[N_EXP];
    const int e = threadIdx.x;

    int tot = 0;
    for (int b = 0; b < NHB; ++b) {
        const int idx = b * N_EXP + e;
        const int c = block_hist[idx];
        block_hist[idx] = tot;          // exclusive prefix per expert
        tot += c;
    }
    gsz[e]    = tot;
    out_gs[e] = (float)tot;             // group_sizes (int values as float)

    float ps = 0.f;                     // deterministic order over blocks
    for (int b = 0; b < NPB; ++b) ps += pblk[(size_t)b * N_EXP + e];
    pt[e] = ps;
    __syncthreads();

    if (e == 0) {
        int off = 0;
        float lb = 0.f;
        for (int x = 0; x < N_EXP; ++x) {
            eoff[x] = off;
            off += gsz[x];
            // f = count/8192 ; p = sum(probs)/8192
            lb += ((float)gsz[x] * (1.0f / 8192.0f)) * (pt[x] * (1.0f / 8192.0f));
        }
        *out_lb = 0.01f * lb;
    }
}

// ---------------------------------------------------------------------------
// Kernel 4: final stable destination per replicated item.
// ---------------------------------------------------------------------------
__global__ void k_dest(const int* __restrict__ sel_idx,
                       const int* __restrict__ lrank,
                       const int* __restrict__ block_hist,
                       const int* __restrict__ eoff,
                       int* __restrict__ dest) {
    const int i = blockIdx.x * blockDim.x + threadIdx.x;
    const int e = sel_idx[i];
    dest[i] = eoff[e] + block_hist[(i / HB) * N_EXP + e] + lrank[i];
}

// ---------------------------------------------------------------------------
// Kernel 5: the bandwidth kernel. One block per token. Load the 8 per-token
// (weight, destination) pairs FIRST (two 16-B vector loads each) so their
// latency overlaps the async DMA, then stage the 8 KB input row via gfx1250
// async global->LDS (ASYNCcnt), s_wait_asynccnt, and write the 8 scaled
// replicas with non-temporal 128-bit stores (output is write-once streaming;
// NT keeps the 192 MB L2 from being thrashed by 512 MB of stores).
// Each lane re-reads exactly the LDS bytes it fetched, so a per-wave
// s_wait_asynccnt suffices (no barrier needed).
// ---------------------------------------------------------------------------
__global__ void k_scatter(const float* __restrict__ x,
                          const float* __restrict__ selw,
                          const int*   __restrict__ dest,
                          float* __restrict__ out) {
    __shared__ __align__(16) float buf[HID];
    const int token = blockIdx.x;
    const int t = threadIdx.x;
    const float* row = x + (size_t)token * HID;

    // ---- metadata first: overlaps with the async row copy below ----
    const v4f w_lo = *(const v4f*)&selw[token * TOPK];       // 32-B aligned
    const v4f w_hi = *(const v4f*)&selw[token * TOPK + 4];
    const v4i d_lo = *(const v4i*)&dest[token * TOPK];
    const v4i d_hi = *(const v4i*)&dest[token * TOPK + 4];
    float wj[TOPK] = { w_lo.x, w_lo.y, w_lo.z, w_lo.w,
                       w_hi.x, w_hi.y, w_hi.z, w_hi.w };
    int   dj[TOPK] = { d_lo.x, d_lo.y, d_lo.z, d_lo.w,
                       d_hi.x, d_hi.y, d_hi.z, d_hi.w };

    // ---- async DMA: global row -> LDS (two b128 per lane) ----
    const unsigned l0 = (unsigned)(uintptr_t)&buf[t * 4];
    const unsigned l1 = (unsigned)(uintptr_t)&buf[1024 + t * 4];
    const float* g0 = row + t * 4;
    const float* g1 = row + 1024 + t * 4;
    asm volatile("global_load_async_to_lds_b128 %0, %1, off"
                 :: "v"(l0), "v"(g0) : "memory");
    asm volatile("global_load_async_to_lds_b128 %0, %1, off"
                 :: "v"(l1), "v"(g1) : "memory");
    asm volatile("s_wait_asynccnt 0" ::: "memory");

    const v4f a = *(const v4f*)&buf[t * 4];
    const v4f b = *(const v4f*)&buf[1024 + t * 4];

    #pragma unroll
    for (int j = 0; j < TOPK; ++j) {
        float* orow = out + (size_t)dj[j] * HID;
        const v4f s0 = a * wj[j];
        const v4f s1 = b * wj[j];
        __builtin_nontemporal_store(s0, (v4f*)(orow + t * 4));
        __builtin_nontemporal_store(s1, (v4f*)(orow + 1024 + t * 4));
    }
}

// ---------------------------------------------------------------------------
extern "C" void kernel_launch(void* const* d_in, const int* in_sizes, int n_in,
                              void* d_out, int out_size, void* d_ws, size_t ws_size,
                              hipStream_t stream) {
    const float* x      = (const float*)d_in[0];   // [8192, 2048]
    const float* logits = (const float*)d_in[1];   // [8192, 64]
    float* out = (float*)d_out;                    // weighted_sorted | group_sizes | lb

    char* ws = (char*)d_ws;
    int*   eoff    = (int*)ws;   ws += N_EXP * sizeof(int);
    int*   gsz     = (int*)ws;   ws += N_EXP * sizeof(int);
    float* sel_w   = (float*)ws; ws += (size_t)N_ITEM * sizeof(float);
    int*   sel_idx = (int*)ws;   ws += (size_t)N_ITEM * sizeof(int);
    int*   lrank   = (int*)ws;   ws += (size_t)N_ITEM * sizeof(int);
    int*   dest    = (int*)ws;   ws += (size_t)N_ITEM * sizeof(int);
    int*   bhist   = (int*)ws;   ws += (size_t)NHB * N_EXP * sizeof(int);
    float* pblk    = (float*)ws; ws += (size_t)NPB * N_EXP * sizeof(float);
    // total scratch ~1.3 MB

    float* out_gs = out + (size_t)N_ITEM * HID;   // 64 group sizes (as float)
    float* out_lb = out_gs + N_EXP;               // scalar lb loss

    k_route  <<<NPB, 256, 0, stream>>>(logits, sel_w, sel_idx, pblk);
    k_hist   <<<NHB, HB, 0, stream>>>(sel_idx, bhist, lrank);
    k_scan   <<<1, N_EXP, 0, stream>>>(bhist, pblk, eoff, gsz, out_gs, out_lb);
    k_dest   <<<N_ITEM / 256, 256, 0, stream>>>(sel_idx, lrank, bhist, eoff, dest);
    k_scatter<<<N_TOK, 256, 0, stream>>>(x, sel_w, dest, out);
}